// YOLOLoss_15556371546343
// MI455X (gfx1250) — compile-verified
//
#include <hip/hip_runtime.h>
#include <math.h>

// ---------------------------------------------------------------------------
// YOLO v1 loss on gfx1250 (MI455X). Memory-bound f32 streaming reduction.
//   kernel 1: one 256-thread block per 256 cells
//             - async-copy (ASYNCcnt / LDS) the 256x30 target tile, b128
//             - coalesced direct loads of the strided prediction channels
//             - per-cell loss, WMMA(16x16x4_f32, ones-A) wave reduction
//             - one partial per block into d_ws
//   kernel 2: single block reduces all partials -> d_out[0]
// ---------------------------------------------------------------------------

#define CELLPX 64.0f
#define IMGPX  448.0f

typedef float v2f  __attribute__((ext_vector_type(2)));
typedef float v8f  __attribute__((ext_vector_type(8)));
typedef float fx4  __attribute__((ext_vector_type(4)));
typedef int   v4i  __attribute__((ext_vector_type(4)));

#if defined(__has_builtin)
#  if __has_builtin(__builtin_amdgcn_global_load_async_to_lds_b128)
#    define HAVE_ASYNC_LDS 1
#  endif
#endif

// Wave32 sum via V_WMMA_F32_16X16X4_F32 with A == ones(16x4):
//   D[m][n] = sum_k B[k][n]  -> with B.x = lane value, B.y = 0 every lane
//   holds colsum(acc_n + acc_{n+16}); 4 xor-shuffles finish the 16-group.
__device__ __forceinline__ float wave_reduce_wmma(float v) {
  v2f a; a.x = 1.0f; a.y = 1.0f;      // entire 16x4 A matrix = 1.0
  v2f b; b.x = v;    b.y = 0.0f;      // rows K=0,K=2 of B carry lane partials
  v8f c = {};
  c = __builtin_amdgcn_wmma_f32_16x16x4_f32(
      /*neg_a=*/false, a, /*neg_b=*/false, b,
      /*c_mod=*/(short)0, c, /*reuse_a=*/false, /*reuse_b=*/false);
  float r = c[0];                     // lane L: acc_{L%16} + acc_{L%16+16}
  r += __shfl_xor(r, 8, 32);
  r += __shfl_xor(r, 4, 32);
  r += __shfl_xor(r, 2, 32);
  r += __shfl_xor(r, 1, 32);
  return r;                           // full wave sum (all lanes)
}

__device__ __forceinline__ void corners(float c0, float c1, float c2, float c3,
                                        float gx, float gy,
                                        float& x1, float& y1,
                                        float& x2, float& y2) {
  float cx = c0 * CELLPX + gx;
  float cy = c1 * CELLPX + gy;
  float w  = c2 * IMGPX;
  float h  = c3 * IMGPX;
  x1 = fminf(fmaxf(cx - 0.5f * w, 0.0f), IMGPX);
  y1 = fminf(fmaxf(cy - 0.5f * h, 0.0f), IMGPX);
  x2 = fminf(fmaxf(cx + 0.5f * w, 0.0f), IMGPX);
  y2 = fminf(fmaxf(cy + 0.5f * h, 0.0f), IMGPX);
}

__device__ __forceinline__ float iou_f(float ax1, float ay1, float ax2, float ay2,
                                       float bx1, float by1, float bx2, float by2) {
  float l  = fmaxf(ax1, bx1);
  float r  = fminf(ax2, bx2);
  float t  = fmaxf(ay1, by1);
  float bo = fminf(ay2, by2);
  bool  m  = (l < r) && (t < bo);
  float inter = (r - l) * (bo - t);
  float uni   = (ax2 - ax1) * (ay2 - ay1) + (bx2 - bx1) * (by2 - by1);
  float den   = uni - inter;
  return m ? (inter / den) : 0.0f;
}

__global__ __launch_bounds__(256)
void yolo_loss_partial(const float* __restrict__ input,
                       const float* __restrict__ target,
                       float* __restrict__ partial,
                       int n_cells) {
  __shared__ float ldsT[256 * 30];    // 30 KB target tile
  __shared__ float wsum[8];

  const int       tid = threadIdx.x;
  const long long n0  = (long long)blockIdx.x * 256;
  const long long rem = (long long)n_cells - n0;
  const int cells_here = (rem < 256) ? (int)rem : 256;
  const int tf = cells_here * 30;     // floats to stage

  const float* gbase = target + n0 * 30;

  if (tf == 7680) {                   // full tile: 1920 x b128, 16B aligned
#if defined(HAVE_ASYNC_LDS)
    for (int i = tid; i < 1920; i += 256) {
      __builtin_amdgcn_global_load_async_to_lds_b128(
          (v4i*)((const fx4*)gbase + i),
          (v4i*)((fx4*)ldsT + i),
          /*offset=*/0, /*cpol=*/0);
    }
#  if __has_builtin(__builtin_amdgcn_s_wait_asynccnt)
    __builtin_amdgcn_s_wait_asynccnt(0);
#  else
    asm volatile("s_wait_asynccnt 0" ::: "memory");
#  endif
#else
    for (int i = tid; i < 1920; i += 256)
      ((fx4*)ldsT)[i] = ((const fx4*)gbase)[i];
#endif
  } else {                            // tail block (not hit for B=16384)
    for (int i = tid; i < tf; i += 256) ldsT[i] = gbase[i];
  }
  __syncthreads();

  float acc = 0.0f;
  const long long n = n0 + tid;
  if (n < (long long)n_cells) {
    const int b      = (int)(n / 49);
    const int cellid = (int)(n - (long long)b * 49);
    const int row    = cellid / 7;
    const int col    = cellid - row * 7;
    const float gx   = (float)col * CELLPX;
    const float gy   = (float)row * CELLPX;

    const float* ip = input + (long long)b * 1470 + cellid;  // chan stride 49
    const float* tp = ldsT + tid * 30;                       // contiguous row

    __builtin_prefetch(ip, 0, 0);     // global_prefetch_b8

    float x[30];
#pragma unroll
    for (int c = 0; c < 30; ++c) x[c] = ip[c * 49];          // coalesced lanes

    float ax1, ay1, ax2, ay2, bx1, by1, bx2, by2;
    float tx1, ty1, tx2, ty2, ux1, uy1, ux2, uy2;
    corners(x[0], x[1], x[2], x[3], gx, gy, ax1, ay1, ax2, ay2);
    corners(x[5], x[6], x[7], x[8], gx, gy, bx1, by1, bx2, by2);
    corners(tp[0], tp[1], tp[2], tp[3], gx, gy, tx1, ty1, tx2, ty2);
    corners(tp[5], tp[6], tp[7], tp[8], gx, gy, ux1, uy1, ux2, uy2);

    float iou1 = iou_f(ax1, ay1, ax2, ay2, tx1, ty1, tx2, ty2);
    float iou2 = iou_f(bx1, by1, bx2, by2, ux1, uy1, ux2, uy2);
    bool  mask = iou1 < iou2;
    float iou  = mask ? iou2 : iou1;

    float s0 = mask ? x[5] : x[0];
    float s1 = mask ? x[6] : x[1];
    float s2 = mask ? x[7] : x[2];
    float s3 = mask ? x[8] : x[3];
    float s4 = mask ? x[9] : x[4];

    float w = (tp[4] == 1.0f) ? 1.0f : 0.0f;

    float d0 = s0 - tp[0], d1 = s1 - tp[1];
    float coord = d0 * d0 + d1 * d1;
    float e0 = sqrtf(s2) - sqrtf(tp[2]);
    float e1 = sqrtf(s3) - sqrtf(tp[3]);
    float size = e0 * e0 + e1 * e1;
    float dc = s4 - iou;
    float conf = dc * dc;
    float noobj = s4 * s4;
    float cls = 0.0f;
#pragma unroll
    for (int c = 10; c < 30; ++c) {
      float d = x[c] - tp[c];
      cls += d * d;
    }

    acc = w * (5.0f * (coord + size) + conf + cls) + 0.5f * (1.0f - w) * noobj;
  }
  // all lanes reconverged here -> EXEC all ones for WMMA
  float wsr = wave_reduce_wmma(acc);
  const int wid = tid >> 5;
  if ((tid & 31) == 0) wsum[wid] = wsr;
  __syncthreads();
  if (tid == 0) {
    float s = 0.0f;
#pragma unroll
    for (int i = 0; i < 8; ++i) s += wsum[i];
    partial[blockIdx.x] = s;
  }
}

__global__ __launch_bounds__(256)
void reduce_partials(const float* __restrict__ partial, int n,
                     float* __restrict__ out) {
  __shared__ float wsum[8];
  float acc = 0.0f;
  for (int i = threadIdx.x; i < n; i += 256) acc += partial[i];
  float wsr = wave_reduce_wmma(acc);
  const int wid = threadIdx.x >> 5;
  if ((threadIdx.x & 31) == 0) wsum[wid] = wsr;
  __syncthreads();
  if (threadIdx.x == 0) {
    float s = 0.0f;
#pragma unroll
    for (int i = 0; i < 8; ++i) s += wsum[i];
    out[0] = s;
  }
}

extern "C" void kernel_launch(void* const* d_in, const int* in_sizes, int n_in,
                              void* d_out, int out_size, void* d_ws, size_t ws_size,
                              hipStream_t stream) {
  const float* input  = (const float*)d_in[0];   // [B,30,7,7]
  const float* target = (const float*)d_in[1];   // [B,7,7,30]
  float*       out    = (float*)d_out;           // scalar
  float*       part   = (float*)d_ws;            // per-block partials

  const int n_cells = in_sizes[1] / 30;          // B*49
  const int blocks  = (n_cells + 255) / 256;     // 3136 for B=16384

  yolo_loss_partial<<<blocks, 256, 0, stream>>>(input, target, part, n_cells);
  reduce_partials<<<1, 256, 0, stream>>>(part, blocks, out);
}